// Attn_89661737271514
// MI455X (gfx1250) — compile-verified
//
#include <hip/hip_runtime.h>
#include <hip/hip_bf16.h>
#include <math.h>

typedef float v2f __attribute__((ext_vector_type(2)));
typedef float v8f __attribute__((ext_vector_type(8)));

#define B_SZ 64
#define L_SZ 1024
#define H_SZ 512

// ---------------------------------------------------------------------------
// Phase 1: v[b][h] = sum_o hidden[b][o] * W[o][h]   (GEMM M=64, N=512, K=512)
// One wave (32 lanes) per 16x16 output tile, K stepped by 4 via
// V_WMMA_F32_16X16X4_F32.  Grid = 4 (M tiles) * 32 (N tiles) = 128 waves.
// A 16x4 layout: lane<16 -> M=lane, K=k+{0,1}; lane>=16 -> M=lane-16, K=k+{2,3}
// B 4x16 layout: mirrored (K = vgpr + 2*half), N striped over lanes 0..15.
// C/D 16x16:     VGPR v, lanes 0..15 -> M=v, lanes 16..31 -> M=v+8.
// ---------------------------------------------------------------------------
__global__ void attn_proj_wmma(const float* __restrict__ hidden,
                               const float* __restrict__ W,
                               float* __restrict__ vbuf) {
    const int tile = blockIdx.x;
    const int m0   = (tile >> 5) << 4;   // 4 M-tiles
    const int n0   = (tile & 31) << 4;   // 32 N-tiles
    const int lane = threadIdx.x & 31;
    const int half = lane >> 4;          // 0: lanes 0-15, 1: lanes 16-31
    const int idx  = lane & 15;

#if defined(__has_builtin) && __has_builtin(__builtin_amdgcn_wmma_f32_16x16x4_f32)
    v8f acc = {};
    const int arow = (m0 + idx) * H_SZ;
    for (int k = 0; k < H_SZ; k += 4) {
        const int ka = k + 2 * half;
        v2f a  = *(const v2f*)(hidden + arow + ka);            // A[m][ka], A[m][ka+1]
        v2f bb;
        bb.x = W[(size_t)ka * H_SZ + n0 + idx];                // B[ka][n]
        bb.y = W[(size_t)(ka + 1) * H_SZ + n0 + idx];          // B[ka+1][n]
        acc = __builtin_amdgcn_wmma_f32_16x16x4_f32(
            /*neg_a=*/false, a, /*neg_b=*/false, bb,
            /*c_mod=*/(short)0, acc, /*reuse_a=*/false, /*reuse_b=*/false);
    }
#pragma unroll
    for (int r = 0; r < 8; ++r) {
        const int m = m0 + r + 8 * half;
        vbuf[m * H_SZ + n0 + idx] = acc[r];
    }
#else
    // Scalar fallback (keeps kernel correct if builtin name/signature differs)
    for (int e = lane; e < 16 * 16; e += 32) {
        const int m = m0 + (e >> 4), n = n0 + (e & 15);
        float s = 0.f;
        for (int k = 0; k < H_SZ; ++k) s = fmaf(hidden[m * H_SZ + k], W[(size_t)k * H_SZ + n], s);
        vbuf[m * H_SZ + n] = s;
    }
#endif
}

// ---------------------------------------------------------------------------
// hbias[b] = dot(hidden[b,:], bias)  -- one wave per batch row
// ---------------------------------------------------------------------------
__global__ void attn_hbias(const float* __restrict__ hidden,
                           const float* __restrict__ bias,
                           float* __restrict__ hbias) {
    const int b    = blockIdx.x;
    const int lane = threadIdx.x & 31;
    float s = 0.f;
    for (int h = lane; h < H_SZ; h += 32)
        s = fmaf(hidden[b * H_SZ + h], bias[h], s);
#pragma unroll
    for (int m = 16; m > 0; m >>= 1) s += __shfl_xor(s, m, 32);
    if (lane == 0) hbias[b] = s;
}

// ---------------------------------------------------------------------------
// Phase 2 (HBM-bound, 128 MB stream):
// energ[b][l] = dot(E[l,b,:], v[b,:]) + hbias[b]
// Grid (16 chunks, 64 batch), 256 threads (8 waves); each wave does 8 rows.
// v[b] (2 KB) staged in LDS; rows read as coalesced float4 (global_load_b128).
// ---------------------------------------------------------------------------
__global__ void attn_energy(const float* __restrict__ enc,
                            const float* __restrict__ vbuf,
                            const float* __restrict__ hbias,
                            float* __restrict__ energ) {
    const int b     = blockIdx.y;
    const int chunk = blockIdx.x;
    __shared__ float vlds[H_SZ];
    for (int i = threadIdx.x; i < H_SZ; i += 256)
        vlds[i] = vbuf[b * H_SZ + i];
    __syncthreads();

    const int wave = threadIdx.x >> 5;
    const int lane = threadIdx.x & 31;
    const float4* v4 = (const float4*)vlds;
    const float  hb  = hbias[b];

    for (int r = 0; r < 8; ++r) {
        const int l = chunk * 64 + wave * 8 + r;
        const float4* row = (const float4*)(enc + ((size_t)l * B_SZ + (size_t)b) * H_SZ);
        if (r + 1 < 8) {  // prefetch next row's cachelines (global_prefetch_b8)
            const float* nrow = enc + ((size_t)(l + 1) * B_SZ + (size_t)b) * H_SZ;
            __builtin_prefetch((const void*)(nrow + lane * 16), 0, 0);
        }
        float s = 0.f;
#pragma unroll
        for (int i = 0; i < 4; ++i) {
            const float4 x  = row[i * 32 + lane];
            const float4 vv = v4[i * 32 + lane];
            s = fmaf(x.x, vv.x, s);
            s = fmaf(x.y, vv.y, s);
            s = fmaf(x.z, vv.z, s);
            s = fmaf(x.w, vv.w, s);
        }
#pragma unroll
        for (int m = 16; m > 0; m >>= 1) s += __shfl_xor(s, m, 32);
        if (lane == 0) energ[b * L_SZ + l] = s + hb;
    }
}

// ---------------------------------------------------------------------------
// Phase 3: softmax over L per batch row; out[b][0][l]
// ---------------------------------------------------------------------------
__global__ void attn_softmax(const float* __restrict__ energ,
                             float* __restrict__ out) {
    const int b = blockIdx.x;
    const int t = threadIdx.x;
    __shared__ float red[256];

    float e[4];
    float mx = -INFINITY;
#pragma unroll
    for (int i = 0; i < 4; ++i) {
        e[i] = energ[b * L_SZ + i * 256 + t];
        mx = fmaxf(mx, e[i]);
    }
    red[t] = mx;
    __syncthreads();
    for (int s = 128; s > 0; s >>= 1) {
        if (t < s) red[t] = fmaxf(red[t], red[t + s]);
        __syncthreads();
    }
    mx = red[0];
    __syncthreads();

    float sum = 0.f;
#pragma unroll
    for (int i = 0; i < 4; ++i) {
        e[i] = __expf(e[i] - mx);
        sum += e[i];
    }
    red[t] = sum;
    __syncthreads();
    for (int s = 128; s > 0; s >>= 1) {
        if (t < s) red[t] += red[t + s];
        __syncthreads();
    }
    const float inv = 1.0f / red[0];
#pragma unroll
    for (int i = 0; i < 4; ++i)
        out[b * L_SZ + i * 256 + t] = e[i] * inv;
}

extern "C" void kernel_launch(void* const* d_in, const int* in_sizes, int n_in,
                              void* d_out, int out_size, void* d_ws, size_t ws_size,
                              hipStream_t stream) {
    const float* hidden = (const float*)d_in[0];  // [1,64,512]
    const float* enc    = (const float*)d_in[1];  // [1024,64,512]
    const float* W      = (const float*)d_in[2];  // [512,512] (o,h)
    const float* bias   = (const float*)d_in[3];  // [512]
    float*       out    = (float*)d_out;          // [64,1,1024]

    float* vbuf  = (float*)d_ws;                  // 64*512 floats
    float* hbias = vbuf + B_SZ * H_SZ;            // 64 floats
    float* energ = hbias + B_SZ;                  // 64*1024 floats

    attn_proj_wmma<<<128, 32, 0, stream>>>(hidden, W, vbuf);
    attn_hbias<<<B_SZ, 32, 0, stream>>>(hidden, bias, hbias);
    attn_energy<<<dim3(16, B_SZ), 256, 0, stream>>>(enc, vbuf, hbias, energ);
    attn_softmax<<<B_SZ, 256, 0, stream>>>(energ, out);
}